// SparseConv1x1_26070451487304
// MI455X (gfx1250) — compile-verified
//
#include <hip/hip_runtime.h>
#include <hip/hip_bf16.h>

// SparseConv1x1 as dense f32 WMMA GEMM on gfx1250, with LDS staging of the
// dense operand via CDNA5 async global->LDS copies.
//
// out[256, 25088] = densify(W_sparse[256,256]) x X[256, 25088]
// Memory floor: 51 MB @ 23.3 TB/s ~= 2.2 us. Each workgroup stages a
// [K=256, N=64] strip of X into LDS exactly once (64 KB), then all 8 waves
// (16 F-tiles total, 2 per wave) compute from LDS -> X read from HBM/L2 1x.

typedef __attribute__((ext_vector_type(2))) float v2f;
typedef __attribute__((ext_vector_type(8))) float v8f;

// Builtin parameter types per hipcc diagnostic: (as1 int*, as3 int*, imm, imm)
typedef __attribute__((address_space(1))) int* gip;  // global
typedef __attribute__((address_space(3))) int* lip;  // LDS

#define F_OUT 256
#define C_IN  256
#define HW    3136          // 56*56
#define NBATCH 8
#define NCOLS (NBATCH * HW) // 25088

#if defined(__has_builtin)
#  if __has_builtin(__builtin_amdgcn_global_load_async_to_lds_b32)
#    define HAVE_ASYNC_LDS 1
#  endif
#endif
#ifndef HAVE_ASYNC_LDS
#  define HAVE_ASYNC_LDS 0
#endif

// ---------------------------------------------------------------------------
// Kernel 1: zero the packed dense weight buffer (256*256 floats = 256 KB)
// ---------------------------------------------------------------------------
__global__ void zero_w_kernel(float* __restrict__ wpk) {
    wpk[blockIdx.x * 256 + threadIdx.x] = 0.0f;
}

// ---------------------------------------------------------------------------
// Kernel 2: scatter COO (row,col,val) into A-matrix-layout packed W.
// A tile (16x4 f32, wave32): lanes 0-15 hold M with VGPR0=K0,VGPR1=K1;
// lanes 16-31 hold K2,K3. Packed so the GEMM's per-k-step A fetch is one
// coalesced global_load_b64 per lane:
//   W[f, c]: s = c>>2, kh = (c>>1)&1, j = c&1
//   lane = (f&15) | (kh<<4),  f_tile = f>>4
//   wpk[ (((s*16 + f_tile)*32 + lane) << 1) | j ] = val
// ---------------------------------------------------------------------------
__global__ void scatter_w_kernel(const float* __restrict__ vals,
                                 const int* __restrict__ rows,
                                 const int* __restrict__ cols,
                                 int nnz, float* __restrict__ wpk) {
    int k = blockIdx.x * 256 + threadIdx.x;
    if (k >= nnz) return;
    int r = rows[k];
    int c = cols[k];
    int s    = c >> 2;
    int kh   = (c >> 1) & 1;
    int j    = c & 1;
    int lane = (r & 15) | (kh << 4);
    int ft   = r >> 4;
    wpk[(((s * 16 + ft) * 32 + lane) << 1) | j] = vals[k];
}

// ---------------------------------------------------------------------------
// Kernel 3: GEMM. grid=(392), block=256 (8 waves).
// Block bx owns columns n0 = bx*64 .. +63 (3136 % 64 == 0, so one batch).
// Stage X[k=0..255, n0..n0+63] into LDS, K-interleaved so that the (K,K+1)
// pair needed per lane is one aligned ds_load_b64 and the 32 lanes cover all
// 64 LDS banks conflict-free:
//     xs[(s*64 + n)*4 + (k&3)]   with s = k>>2   (16384 floats = 64 KB)
// Wave w computes F-tiles {2w, 2w+1} (all 256 rows covered by the block),
// sharing each B fetch across both F-tiles (halves LDS read traffic).
// ---------------------------------------------------------------------------
__global__ __launch_bounds__(256) void spmm_wmma_kernel(
        const float* __restrict__ x,      // [B, C, HW] = X[C, N] per batch
        const float* __restrict__ wpk,    // packed dense W, A layout
        float* __restrict__ out) {        // [B, F, HW]
    __shared__ float xs[16384];           // 64 KB strip

    const int t   = threadIdx.x;
    const int n0  = blockIdx.x * 64;
    const int b   = n0 / HW;
    const int hw0 = n0 % HW;

    // ---- stage strip into LDS (async global->LDS, ASYNCcnt-tracked) ----
    const float* xg = x + (size_t)b * C_IN * HW + hw0;
    #pragma unroll 8
    for (int i = 0; i < 64; ++i) {
        const int e = i * 256 + t;              // 0..16383
        const int k = e >> 6;                   // channel 0..255
        const int n = e & 63;                   // col in strip
        const int lidx = ((k >> 2) << 8) + (n << 2) + (k & 3);
        const int goff = k * HW + n;
#if HAVE_ASYNC_LDS
        __builtin_amdgcn_global_load_async_to_lds_b32(
            (gip)(xg + goff), (lip)(xs + lidx), 0, 0);
#else
        xs[lidx] = xg[goff];
#endif
    }
#if HAVE_ASYNC_LDS
#  if __has_builtin(__builtin_amdgcn_s_wait_asynccnt)
    __builtin_amdgcn_s_wait_asynccnt(0);
#  else
    asm volatile("s_wait_asynccnt 0x0" ::: "memory");
#  endif
#endif
    __syncthreads();

    // ---- compute: K loop of 64 steps of V_WMMA_F32_16X16X4_F32 ----
    const int lane   = t & 31;
    const int w      = t >> 5;                  // wave 0..7
    const int laneN  = lane & 15;
    const int laneKh = (lane >> 4) << 1;        // 0 or 2

    const v2f* ap = (const v2f*)wpk + (2 * w) * 32 + lane;  // f-tile 2w
    const float* xb = xs + (laneN << 2) + laneKh;

    v8f acc[2][4] = {};

    #pragma unroll 2
    for (int s = 0; s < 64; ++s) {
        v2f av[2];
        av[0] = ap[s * 512];                    // f-tile 2w   (global b64)
        av[1] = ap[s * 512 + 32];               // f-tile 2w+1
        const float* xk = xb + s * 256;
        v2f bv[4];
        bv[0] = *(const v2f*)(xk);              // ds_load_b64, bank-clean
        bv[1] = *(const v2f*)(xk + 64);
        bv[2] = *(const v2f*)(xk + 128);
        bv[3] = *(const v2f*)(xk + 192);
        #pragma unroll
        for (int fi = 0; fi < 2; ++fi)
            #pragma unroll
            for (int a = 0; a < 4; ++a)
                acc[fi][a] = __builtin_amdgcn_wmma_f32_16x16x4_f32(
                    false, av[fi], false, bv[a],
                    (short)0, acc[fi][a], false, false);
    }

    // ---- store: C/D layout VGPR v -> M = v | v+8 by lane half, N = lane&15
    const int hi8 = (lane >> 4) * 8;
    float* op = out + (size_t)b * F_OUT * HW + hw0 + laneN;
    #pragma unroll
    for (int fi = 0; fi < 2; ++fi) {
        const int fbase = (2 * w + fi) * 16 + hi8;
        #pragma unroll
        for (int v = 0; v < 8; ++v) {
            float* row = op + (size_t)(fbase + v) * HW;
            row[0]  = acc[fi][0][v];
            row[16] = acc[fi][1][v];
            row[32] = acc[fi][2][v];
            row[48] = acc[fi][3][v];
        }
    }
}

// ---------------------------------------------------------------------------
extern "C" void kernel_launch(void* const* d_in, const int* in_sizes, int n_in,
                              void* d_out, int out_size, void* d_ws, size_t ws_size,
                              hipStream_t stream) {
    const float* x    = (const float*)d_in[0];   // inputs [8,256,56,56] f32
    const float* vals = (const float*)d_in[1];   // values [nnz] f32
    const int*   rows = (const int*)d_in[2];     // row_ids [nnz] i32
    const int*   cols = (const int*)d_in[3];     // col_ids [nnz] i32
    const int nnz = in_sizes[1];

    float* wpk = (float*)d_ws;                   // 256 KB packed dense W

    zero_w_kernel<<<256, 256, 0, stream>>>(wpk);
    scatter_w_kernel<<<(nnz + 255) / 256, 256, 0, stream>>>(vals, rows, cols, nnz, wpk);

    spmm_wmma_kernel<<<392, 256, 0, stream>>>(x, wpk, (float*)d_out);
}